// SplitEmbedding_4063039062251
// MI455X (gfx1250) — compile-verified
//
#include <hip/hip_runtime.h>
#include <stdint.h>

// Split embedding gather for MI455X (gfx1250).
// out[t, :] = (id2row[ids[t]] >= 0) ? train_weight[id2row[ids[t]], :]
//                                   : base_weight[ids[t], :]
// Pure bandwidth problem: ~201 MB moved -> ~8.6 us floor at 23.3 TB/s.
// Row data moves via gfx1250 async global->LDS->global copies (ASYNCcnt),
// never transiting the VGPR file. Output stores are non-temporal so the
// write-once 100 MB output does not evict base_weight (154 MB) from the
// 192 MB L2, keeping duplicate-token gathers L2-resident.

#define EMBED_DIM 768
#define ROW_BYTES (EMBED_DIM * 4)     // 3072 bytes per row
#define WAVES_PER_BLOCK 8
#define THREADS_PER_BLOCK (WAVES_PER_BLOCK * 32)
#define TUNE_CAP 4096                 // LDS staging capacity for tune_ids

// ---------------------------------------------------------------------------
// Kernel 1: id2row[v] = last index i with tune_ids[i] == v, else -1.
// tune_ids is sorted, so upper_bound-1 implements "later duplicates win"
// with no atomics. tune_ids (4 KB) is staged in LDS so the 10-step binary
// search is dependent ds_loads instead of dependent global round-trips.
// ---------------------------------------------------------------------------
__global__ void build_id2row(const int* __restrict__ tune_ids,
                             int* __restrict__ id2row,
                             int vocab, int ntune) {
  __shared__ int s_tune[TUNE_CAP];
  const bool staged = (ntune <= TUNE_CAP);
  if (staged) {
    for (int i = threadIdx.x; i < ntune; i += blockDim.x)
      s_tune[i] = tune_ids[i];
    __syncthreads();
  }

  int v = blockIdx.x * blockDim.x + threadIdx.x;
  if (v >= vocab) return;

  const int* t = staged ? s_tune : tune_ids;
  int lo = 0, hi = ntune;                 // upper_bound(v)
  while (lo < hi) {
    int mid = (lo + hi) >> 1;
    if (t[mid] <= v) lo = mid + 1; else hi = mid;
  }
  int row = -1;
  if (lo > 0 && t[lo - 1] == v) row = lo - 1;
  id2row[v] = row;
}

// ---------------------------------------------------------------------------
// Kernel 2: one wave copies one 3072-byte row via async global->LDS->global.
// Each lane moves 16 B per instruction (512 B/wave/instruction); the
// instruction offset advances BOTH the global and LDS address, per the ISA
// pseudocode for GLOBAL_LOAD_ASYNC_TO_LDS / GLOBAL_STORE_ASYNC_FROM_LDS.
// ---------------------------------------------------------------------------
__global__ __launch_bounds__(THREADS_PER_BLOCK)
void gather_rows(const int* __restrict__ ids,
                 const int* __restrict__ id2row,
                 const float* __restrict__ base_w,
                 const float* __restrict__ train_w,
                 float* __restrict__ out,
                 int nrows) {
  __shared__ __align__(16) unsigned char smem[WAVES_PER_BLOCK * ROW_BYTES]; // 24 KB

  const int lane = threadIdx.x & 31;
  const int wib  = threadIdx.x >> 5;                    // wave index in block
  const int row  = blockIdx.x * WAVES_PER_BLOCK + wib;  // token index
  if (row >= nrows) return;                             // wave-uniform guard

  const int tok  = ids[row];
  const int trow = id2row[tok];
  const float* src = (trow >= 0) ? (train_w + (size_t)trow * EMBED_DIM)
                                 : (base_w  + (size_t)tok  * EMBED_DIM);

  uint64_t gsrc = (uint64_t)(uintptr_t)src + (uint32_t)(lane * 16);
  uint64_t gdst = (uint64_t)(uintptr_t)(out + (size_t)row * EMBED_DIM)
                  + (uint32_t)(lane * 16);
  // Generic LDS pointer: low 32 bits are the workgroup-relative LDS byte addr.
  uint32_t lds  = (uint32_t)(uintptr_t)&smem[0]
                  + (uint32_t)(wib * ROW_BYTES + lane * 16);

  // --- 6 async b128 loads: global -> LDS (3072 B per wave) ---
  asm volatile(
      "global_load_async_to_lds_b128 %0, %1, off\n\t"
      "global_load_async_to_lds_b128 %0, %1, off offset:512\n\t"
      "global_load_async_to_lds_b128 %0, %1, off offset:1024\n\t"
      "global_load_async_to_lds_b128 %0, %1, off offset:1536\n\t"
      "global_load_async_to_lds_b128 %0, %1, off offset:2048\n\t"
      "global_load_async_to_lds_b128 %0, %1, off offset:2560"
      :
      : "v"(lds), "v"(gsrc)
      : "memory");

  // Async loads and stores complete out of order w.r.t. each other: drain
  // the loads before the stores read the staged row from LDS.
  asm volatile("s_wait_asynccnt 0" ::: "memory");

  // --- 6 async b128 stores: LDS -> global, non-temporal (write-once) ---
  asm volatile(
      "global_store_async_from_lds_b128 %0, %1, off th:TH_STORE_NT\n\t"
      "global_store_async_from_lds_b128 %0, %1, off offset:512 th:TH_STORE_NT\n\t"
      "global_store_async_from_lds_b128 %0, %1, off offset:1024 th:TH_STORE_NT\n\t"
      "global_store_async_from_lds_b128 %0, %1, off offset:1536 th:TH_STORE_NT\n\t"
      "global_store_async_from_lds_b128 %0, %1, off offset:2048 th:TH_STORE_NT\n\t"
      "global_store_async_from_lds_b128 %0, %1, off offset:2560 th:TH_STORE_NT"
      :
      : "v"(gdst), "v"(lds)
      : "memory");

  // S_ENDPGM performs an implicit wait-idle; explicit drain for clarity.
  asm volatile("s_wait_asynccnt 0" ::: "memory");
}

// ---------------------------------------------------------------------------
// Launch
// ---------------------------------------------------------------------------
extern "C" void kernel_launch(void* const* d_in, const int* in_sizes, int n_in,
                              void* d_out, int out_size, void* d_ws, size_t ws_size,
                              hipStream_t stream) {
  const int*   input_ids = (const int*)d_in[0];   // [8*4096]
  const int*   tune_ids  = (const int*)d_in[1];   // [1024] sorted
  const float* base_w    = (const float*)d_in[2]; // [VOCAB,768]
  const float* train_w   = (const float*)d_in[3]; // [NUM_TUNE,768]
  float*       out       = (float*)d_out;

  const int nrows = in_sizes[0];
  const int ntune = in_sizes[1];
  const int vocab = in_sizes[2] / EMBED_DIM;

  int* id2row = (int*)d_ws;                       // vocab ints (~201 KB)

  // 1) deterministic id -> train-row map (rebuilt every call).
  build_id2row<<<(vocab + 255) / 256, 256, 0, stream>>>(tune_ids, id2row,
                                                        vocab, ntune);

  // 2) async-DMA row gather: one wave per row.
  const int blocks = (nrows + WAVES_PER_BLOCK - 1) / WAVES_PER_BLOCK;
  gather_rows<<<blocks, THREADS_PER_BLOCK, 0, stream>>>(
      input_ids, id2row, base_w, train_w, out, nrows);
}